// DensityRPN_24816321036375
// MI455X (gfx1250) — compile-verified
//
#include <hip/hip_runtime.h>
#include <cstdint>
#include <cstddef>

// ---------------- constants matching the reference ----------------
#define PREV_N   6000
#define POST_N   1000
#define NWORDS   94            // ceil(6000/64)
#define CAP      8192          // compaction / bitonic sort capacity (pow2 >= PREV_N)
#define NTOT     261888        // total anchors over 5 FPN levels
#define NEGF     (-1000000000.0f)
#define NMS_THR_F 0.7f

// =================================================================
// 1) Fused per-level decode: box decode + clip + softmax/sigmoid +
//    min-size mask -> sortable u32 key.  Pure streaming (HBM bound).
// =================================================================
__global__ void decode_level_k(const float* __restrict__ off, const float* __restrict__ cls,
                               const float* __restrict__ den, const float* __restrict__ anc,
                               const float* __restrict__ im_info, int H, int W, int base,
                               float* __restrict__ boxes, float* __restrict__ dens,
                               unsigned* __restrict__ keys)
{
    const int n = H * W * 3;
    int g = blockIdx.x * blockDim.x + threadIdx.x;
    if (g >= n) return;
    const int hw  = H * W;
    const int pos = g / 3;
    const int a   = g - pos * 3;

    float dx = off[(a * 4 + 0) * hw + pos];
    float dy = off[(a * 4 + 1) * hw + pos];
    float dw = off[(a * 4 + 2) * hw + pos];
    float dh = off[(a * 4 + 3) * hw + pos];
    float c0 = cls[(a * 2 + 0) * hw + pos];
    float c1 = cls[(a * 2 + 1) * hw + pos];
    float dv = den[a * hw + pos];

    const float4 an = reinterpret_cast<const float4*>(anc)[g];
    float w  = an.z - an.x + 1.f;
    float h  = an.w - an.y + 1.f;
    float cx = an.x + 0.5f * w;
    float cy = an.y + 0.5f * h;
    float pcx = dx * w + cx, pcy = dy * h + cy;
    float pw = expf(dw) * w, ph = expf(dh) * h;
    float x1 = pcx - 0.5f * (pw - 1.f);
    float y1 = pcy - 0.5f * (ph - 1.f);
    float x2 = pcx + 0.5f * (pw - 1.f);
    float y2 = pcy + 0.5f * (ph - 1.f);

    float im_h = im_info[0], im_w = im_info[1], scl = im_info[2];
    x1 = fminf(fmaxf(x1, 0.f), im_w - 1.f);
    y1 = fminf(fmaxf(y1, 0.f), im_h - 1.f);
    x2 = fminf(fmaxf(x2, 0.f), im_w - 1.f);
    y2 = fminf(fmaxf(y2, 0.f), im_h - 1.f);

    float bw = x2 - x1 + 1.f, bh = y2 - y1 + 1.f;
    float minsz = 2.0f * scl;
    bool valid = (bw >= minsz) && (bh >= minsz);

    // softmax over 2 classes, take class 1 == sigmoid(c1 - c0)
    float score = 1.f / (1.f + expf(c0 - c1));
    float s = valid ? score : NEGF;

    // monotonic float -> u32 key (descending score == descending key)
    unsigned bits = __float_as_uint(s);
    unsigned key  = bits ^ ((bits & 0x80000000u) ? 0xFFFFFFFFu : 0x80000000u);

    int gi = base + g;
    reinterpret_cast<float4*>(boxes)[gi] = make_float4(x1, y1, x2, y2);
    dens[gi] = 1.f / (1.f + expf(-dv));
    keys[gi] = key;
}

// =================================================================
// 2) Exact top-PREV_N threshold via 4-pass MSB radix select
//    state[0]=prefix  state[1]=remaining rank K  state[2]=T  state[3]=counter
// =================================================================
__global__ void init_state_k(unsigned* __restrict__ bins, unsigned* __restrict__ state)
{
    int t = threadIdx.x;
    if (t < 256) bins[t] = 0u;
    if (t == 0) { state[0] = 0u; state[1] = (unsigned)PREV_N; state[2] = 0u; state[3] = 0u; }
}

__global__ void hist_pass_k(const unsigned* __restrict__ keys, int n, int pass,
                            unsigned* __restrict__ bins, const unsigned* __restrict__ state)
{
    __shared__ unsigned lb[256];
    if (threadIdx.x < 256) lb[threadIdx.x] = 0u;
    __syncthreads();
    const unsigned prefix = state[0];
    const unsigned himask = pass ? (0xFFFFFFFFu << (32 - 8 * pass)) : 0u;
    const int shift = 24 - 8 * pass;
    for (int g = blockIdx.x * blockDim.x + threadIdx.x; g < n; g += gridDim.x * blockDim.x) {
        unsigned k = keys[g];
        if ((k & himask) == (prefix & himask))
            atomicAdd(&lb[(k >> shift) & 0xFFu], 1u);
    }
    __syncthreads();
    if (threadIdx.x < 256 && lb[threadIdx.x]) atomicAdd(&bins[threadIdx.x], lb[threadIdx.x]);
}

__global__ void scan_pass_k(unsigned* __restrict__ bins, unsigned* __restrict__ state, int pass)
{
    __shared__ unsigned sb[256];
    int t = threadIdx.x;
    sb[t] = bins[t];
    __syncthreads();
    if (t == 0) {
        unsigned K = state[1], prefix = state[0];
        const int shift = 24 - 8 * pass;
        unsigned cum = 0;
        for (int b = 255; b >= 0; --b) {
            unsigned c = sb[b];
            if (cum + c >= K) { prefix |= ((unsigned)b) << shift; K -= cum; break; }
            cum += c;
        }
        state[0] = prefix; state[1] = K;
        if (pass == 3) { state[2] = prefix; state[3] = 0u; }   // final threshold T, reset counter
    }
    bins[t] = 0u;   // each thread already copied its own bin
}

__global__ void compact_k(const unsigned* __restrict__ keys, int n,
                          unsigned* __restrict__ state, unsigned long long* __restrict__ list)
{
    const unsigned T = state[2];
    for (int g = blockIdx.x * blockDim.x + threadIdx.x; g < n; g += gridDim.x * blockDim.x) {
        unsigned k = keys[g];
        if (k >= T) {
            unsigned p = atomicAdd(&state[3], 1u);
            if (p < CAP)
                list[p] = (((unsigned long long)(~k)) << 32) | (unsigned long long)(unsigned)g;
        }
    }
}

// single-block bitonic sort of CAP u64 values ascending:
// value = (~key)<<32 | idx  =>  score descending, index ascending (JAX top_k tie order)
__global__ __launch_bounds__(1024) void sort_top_k(const unsigned long long* __restrict__ list,
                                                   const unsigned* __restrict__ state,
                                                   unsigned long long* __restrict__ top)
{
    extern __shared__ unsigned long long s[];   // CAP * 8 = 64 KB dynamic LDS
    unsigned cnt = state[3];
    if (cnt > CAP) cnt = CAP;
    for (int i = threadIdx.x; i < CAP; i += 1024)
        s[i] = (i < (int)cnt) ? list[i] : 0xFFFFFFFFFFFFFFFFull;
    __syncthreads();
    for (unsigned k = 2; k <= CAP; k <<= 1) {
        for (unsigned j = k >> 1; j > 0; j >>= 1) {
            for (unsigned i = threadIdx.x; i < CAP; i += 1024) {
                unsigned ixj = i ^ j;
                if (ixj > i) {
                    unsigned long long a = s[i], b = s[ixj];
                    bool up = ((i & k) == 0);
                    if ((a > b) == up) { s[i] = b; s[ixj] = a; }
                }
            }
            __syncthreads();
        }
    }
    for (int t = threadIdx.x; t < PREV_N; t += 1024) top[t] = s[t];
}

// =================================================================
// 3) Gather top boxes/densities + validity flag
// =================================================================
__global__ void gather_top_k(const unsigned long long* __restrict__ top,
                             const float* __restrict__ boxes, const float* __restrict__ dens,
                             float* __restrict__ tb, float* __restrict__ td,
                             unsigned* __restrict__ tvalid)
{
    int t = blockIdx.x * blockDim.x + threadIdx.x;
    if (t >= PREV_N) return;
    unsigned long long c = top[t];
    if (c == 0xFFFFFFFFFFFFFFFFull) {        // padding slot
        reinterpret_cast<float4*>(tb)[t] = make_float4(0.f, 0.f, 0.f, 0.f);
        td[t] = 0.f; tvalid[t] = 0u;
        return;
    }
    unsigned key  = ~((unsigned)(c >> 32));
    unsigned idx  = (unsigned)c;
    unsigned bits = (key & 0x80000000u) ? (key ^ 0x80000000u) : ~key;  // key -> float bits
    float sc = __uint_as_float(bits);
    reinterpret_cast<float4*>(tb)[t] = reinterpret_cast<const float4*>(boxes)[idx];
    td[t] = dens[idx];
    tvalid[t] = (sc > NEGF * 0.5f) ? 1u : 0u;    // keep-init: top_sc > NEG/2
}

// =================================================================
// 4) IoU suppression-mask matrix: mask[i][w] bit c set iff
//    j = w*64+c > i, j < PREV_N, IoU(i,j) > 0.7.
//    Column tile staged in LDS via CDNA5 async load-to-LDS.
// =================================================================
__global__ __launch_bounds__(64) void nms_mask_k(const float4* __restrict__ tb4,
                                                 unsigned long long* __restrict__ mask)
{
    if (blockIdx.x < blockIdx.y) return;         // sub-diagonal words are never read
    __shared__ float4 jb[64];
    const int lane  = threadIdx.x;
    const int jbase = blockIdx.x * 64;

    // --- async copy of 64 column boxes (1 KB) into LDS (gfx1250 path) ---
    {
        unsigned ldsaddr = (unsigned)(size_t)(&jb[lane]);     // low 32 bits of flat LDS ptr = LDS offset
        const float4* src = tb4 + (jbase + lane);             // stays inside workspace (td follows tb)
        asm volatile("global_load_async_to_lds_b128 %0, %1, off"
                     :: "v"(ldsaddr), "v"(src) : "memory");
    }
#if __has_builtin(__builtin_amdgcn_s_wait_asynccnt)
    __builtin_amdgcn_s_wait_asynccnt(0);
#else
    asm volatile("s_wait_asynccnt 0" ::: "memory");
#endif
    __syncthreads();

    const int i = blockIdx.y * 64 + lane;
    const float4 bi = tb4[(i < PREV_N) ? i : 0];
    const float areai = (bi.z - bi.x + 1.f) * (bi.w - bi.y + 1.f);
    unsigned long long m = 0ull;
#pragma unroll 4
    for (int c = 0; c < 64; ++c) {
        int j = jbase + c;
        float4 bj = jb[c];
        float ix1 = fmaxf(bi.x, bj.x), iy1 = fmaxf(bi.y, bj.y);
        float ix2 = fminf(bi.z, bj.z), iy2 = fminf(bi.w, bj.w);
        float iw = fmaxf(ix2 - ix1 + 1.f, 0.f);
        float ih = fmaxf(iy2 - iy1 + 1.f, 0.f);
        float inter = iw * ih;
        float areaj = (bj.z - bj.x + 1.f) * (bj.w - bj.y + 1.f);
        float iou = inter / (areai + areaj - inter);
        bool sup = (j > i) && (j < PREV_N) && (iou > NMS_THR_F);
        m |= (sup ? 1ull : 0ull) << c;
    }
    if (i < PREV_N) mask[(size_t)i * NWORDS + blockIdx.x] = m;
}

// =================================================================
// 5) Sequential greedy-NMS reduction (word-blocked: 94 serial steps,
//    not 6000) + stable compaction + output rows [0, box, density].
// =================================================================
__global__ void nms_seq_out_k(const unsigned long long* __restrict__ mask,
                              const unsigned* __restrict__ tvalid,
                              const float* __restrict__ tb, const float* __restrict__ td,
                              float* __restrict__ out)
{
    __shared__ unsigned long long removed[NWORDS];
    __shared__ unsigned long long colbuf[64];
    __shared__ unsigned char keepf[PREV_N];
    __shared__ int keptList[64];
    __shared__ int nK;
    __shared__ int psum[257];
    const int tid = threadIdx.x;
    const int nt  = blockDim.x;   // 256

    for (int w = tid; w < NWORDS; w += nt) {
        unsigned long long r = 0ull;
        for (int b = 0; b < 64; ++b) {
            int i = w * 64 + b;
            if (i >= PREV_N || tvalid[i] == 0u) r |= (1ull << b);
        }
        removed[w] = r;
    }
    for (int i = tid; i < PREV_N; i += nt) keepf[i] = 0;
    __syncthreads();

    for (int w = 0; w < NWORDS; ++w) {
        if (tid < 64) {
            int i = w * 64 + tid;
            colbuf[tid] = (i < PREV_N) ? mask[(size_t)i * NWORDS + w] : 0ull;
        }
        __syncthreads();
        if (tid == 0) {
            unsigned long long r = removed[w];
            int cnt = 0;
            for (int b = 0; b < 64 && (w * 64 + b) < PREV_N; ++b) {
                if (!((r >> b) & 1ull)) {
                    int i = w * 64 + b;
                    keepf[i] = 1;
                    keptList[cnt++] = i;
                    r |= colbuf[b];          // within-word suppression (bits only > b)
                }
            }
            removed[w] = r;
            nK = cnt;
        }
        __syncthreads();
        const int cnt = nK;
        for (int k = 0; k < cnt; ++k) {      // cross-word suppression in parallel
            int i = keptList[k];
            for (int ww = w + 1 + tid; ww < NWORDS; ww += nt)
                removed[ww] |= mask[(size_t)i * NWORDS + ww];
        }
        __syncthreads();
    }

    // zero output (harness poisons it), then stable-compact kept rows
    for (int r = tid; r < POST_N * 6; r += nt) out[r] = 0.f;

    const int CHUNK = (PREV_N + 255) / 256;  // 24
    int base_i = tid * CHUNK;
    int cnt = 0;
    for (int k = 0; k < CHUNK; ++k) {
        int i = base_i + k;
        if (i < PREV_N && keepf[i]) cnt++;
    }
    psum[tid + 1] = cnt;
    __syncthreads();
    if (tid == 0) { psum[0] = 0; for (int t = 0; t < 256; ++t) psum[t + 1] += psum[t]; }
    __syncthreads();
    int rank = psum[tid];
    for (int k = 0; k < CHUNK; ++k) {
        int i = base_i + k;
        if (i < PREV_N && keepf[i]) {
            if (rank < POST_N) {
                out[rank * 6 + 0] = 0.f;
                out[rank * 6 + 1] = tb[i * 4 + 0];
                out[rank * 6 + 2] = tb[i * 4 + 1];
                out[rank * 6 + 3] = tb[i * 4 + 2];
                out[rank * 6 + 4] = tb[i * 4 + 3];
                out[rank * 6 + 5] = td[i];
            }
            rank++;
        }
    }
}

// =================================================================
// host-side launch
// =================================================================
extern "C" void kernel_launch(void* const* d_in, const int* in_sizes, int n_in,
                              void* d_out, int out_size, void* d_ws, size_t ws_size,
                              hipStream_t stream)
{
    (void)out_size; (void)ws_size;
    static const int LH[5] = {256, 128, 64, 32, 16};
    static const int LW[5] = {256, 128, 64, 32, 16};
    int bases[5]; { int b = 0; for (int l = 0; l < 5; ++l) { bases[l] = b; b += LH[l] * LW[l] * 3; } }

    // input order auto-detection: setup_inputs() dict order interleaves
    // (offsets, cls, density, anchors) per level; signature order groups them.
    bool dictOrder = (n_in == 21) && (in_sizes[1] == 6 * 256 * 256);
    const float* offp[5]; const float* clsp[5]; const float* denp[5]; const float* ancp[5];
    for (int l = 0; l < 5; ++l) {
        if (dictOrder) {
            offp[l] = (const float*)d_in[4 * l + 0];
            clsp[l] = (const float*)d_in[4 * l + 1];
            denp[l] = (const float*)d_in[4 * l + 2];
            ancp[l] = (const float*)d_in[4 * l + 3];
        } else {
            offp[l] = (const float*)d_in[l];
            clsp[l] = (const float*)d_in[5 + l];
            denp[l] = (const float*)d_in[10 + l];
            ancp[l] = (const float*)d_in[15 + l];
        }
    }
    const float* im_info = (const float*)d_in[20];

    // ---- workspace layout (all 16B aligned), ~11.1 MB total ----
    char* w = (char*)d_ws;
    size_t o = 0;
    unsigned long long* d_mask = (unsigned long long*)(w + o); o += (size_t)PREV_N * NWORDS * 8;
    unsigned long long* d_list = (unsigned long long*)(w + o); o += (size_t)CAP * 8;
    unsigned long long* d_top  = (unsigned long long*)(w + o); o += (size_t)PREV_N * 8;
    unsigned* d_keys  = (unsigned*)(w + o); o += (size_t)NTOT * 4;
    float*    d_boxes = (float*)(w + o);    o += (size_t)NTOT * 16;
    float*    d_dens  = (float*)(w + o);    o += (size_t)NTOT * 4;
    float*    d_tb    = (float*)(w + o);    o += (size_t)PREV_N * 16;
    float*    d_td    = (float*)(w + o);    o += (size_t)PREV_N * 4;
    unsigned* d_tvld  = (unsigned*)(w + o); o += (size_t)PREV_N * 4;
    unsigned* d_bins  = (unsigned*)(w + o); o += 256 * 4;
    unsigned* d_state = (unsigned*)(w + o); o += 64;

    float* out = (float*)d_out;

    init_state_k<<<1, 256, 0, stream>>>(d_bins, d_state);

    for (int l = 0; l < 5; ++l) {
        int n = LH[l] * LW[l] * 3;
        decode_level_k<<<(n + 255) / 256, 256, 0, stream>>>(
            offp[l], clsp[l], denp[l], ancp[l], im_info,
            LH[l], LW[l], bases[l], d_boxes, d_dens, d_keys);
    }

    for (int p = 0; p < 4; ++p) {
        hist_pass_k<<<512, 256, 0, stream>>>(d_keys, NTOT, p, d_bins, d_state);
        scan_pass_k<<<1, 256, 0, stream>>>(d_bins, d_state, p);
    }

    compact_k<<<512, 256, 0, stream>>>(d_keys, NTOT, d_state, d_list);
    sort_top_k<<<1, 1024, (size_t)CAP * 8, stream>>>(d_list, d_state, d_top);
    gather_top_k<<<(PREV_N + 255) / 256, 256, 0, stream>>>(d_top, d_boxes, d_dens, d_tb, d_td, d_tvld);

    nms_mask_k<<<dim3(NWORDS, NWORDS), 64, 0, stream>>>(
        reinterpret_cast<const float4*>(d_tb), d_mask);

    nms_seq_out_k<<<1, 256, 0, stream>>>(d_mask, d_tvld, d_tb, d_td, out);
}